// BaseEnergyFormPhysics_67568425501088
// MI455X (gfx1250) — compile-verified
//
#include <hip/hip_runtime.h>

typedef __attribute__((ext_vector_type(2))) float v2f;
typedef __attribute__((ext_vector_type(4))) float v4f;
typedef __attribute__((ext_vector_type(8))) float v8f;
typedef __attribute__((ext_vector_type(4))) int   v4i;

#define WAVE 32
#define BLOCK 256
#define NBLOCKS 2048

// ---------------------------------------------------------------------------
// Wave-level sum of (acc0 + acc1) across all 32 lanes using the matrix pipe.
// A (16x4 f32) = the 64 partials (2 VGPRs x 32 lanes), B = ones, C = 0.
// D[m][n] = acc0[m] + acc1[m] + acc0[m+16] + acc1[m+16]   (replicated over n)
// Lane L in [0,15]:  D vgpr v holds rowsum[v]    (M = v,   N = L)
// Lane L in [16,31]: D vgpr v holds rowsum[v+8]  (M = v+8, N = L-16)
// pairwise sum of the 8 D components + shfl_xor(16) -> full wave sum.
// Requires EXEC == all ones (callers run with full, non-divergent blocks).
// ---------------------------------------------------------------------------
__device__ __forceinline__ float wave_sum_wmma(float acc0, float acc1) {
    v2f a; a[0] = acc0; a[1] = acc1;
    v2f b; b[0] = 1.0f; b[1] = 1.0f;
    v8f c = {};
    v8f d = __builtin_amdgcn_wmma_f32_16x16x4_f32(
        /*neg_a=*/false, a, /*neg_b=*/false, b,
        /*c_mod=*/(short)0, c, /*reuse_a=*/false, /*reuse_b=*/false);
    float s0 = d[0] + d[1];
    float s1 = d[2] + d[3];
    float s2 = d[4] + d[5];
    float s3 = d[6] + d[7];
    float s  = (s0 + s1) + (s2 + s3);
    s += __shfl_xor(s, 16, WAVE);   // rowsums[0..7] + rowsums[8..15]
    return s;
}

// ---------------------------------------------------------------------------
// Per-element TET4 linear-elasticity energy (1-pt quadrature), MU=0.5, LAM=1.
// ---------------------------------------------------------------------------
__device__ __forceinline__ float elem_energy(int e,
                                             const float* __restrict__ coords,
                                             const float* __restrict__ us,
                                             const int*   __restrict__ conns) {
    // Streaming index load: non-temporal so L2 stays owned by coords/us.
    v4i n = __builtin_nontemporal_load((const v4i*)conns + e);

    const float* p0 = coords + 3u * (unsigned)n[0];
    const float* p1 = coords + 3u * (unsigned)n[1];
    const float* p2 = coords + 3u * (unsigned)n[2];
    const float* p3 = coords + 3u * (unsigned)n[3];
    float x0 = p0[0], y0 = p0[1], z0 = p0[2];

    // Jacobian columns = edge vectors (dN is +-1/0)
    float ax = p1[0] - x0, ay = p1[1] - y0, az = p1[2] - z0;
    float bx = p2[0] - x0, by = p2[1] - y0, bz = p2[2] - z0;
    float cx = p3[0] - x0, cy = p3[1] - y0, cz = p3[2] - z0;

    // inv(J) rows: r0 = b x c / det, r1 = c x a / det, r2 = a x b / det
    float r0x = by*cz - bz*cy, r0y = bz*cx - bx*cz, r0z = bx*cy - by*cx;
    float r1x = cy*az - cz*ay, r1y = cz*ax - cx*az, r1z = cx*ay - cy*ax;
    float r2x = ay*bz - az*by, r2y = az*bx - ax*bz, r2z = ax*by - ay*bx;

    float det = ax*r0x + ay*r0y + az*r0z;
    float inv = 1.0f / det;                 // inf/NaN propagates like reference
    r0x *= inv; r0y *= inv; r0z *= inv;
    r1x *= inv; r1y *= inv; r1z *= inv;
    r2x *= inv; r2y *= inv; r2z *= inv;

    const float* q0 = us + 3u * (unsigned)n[0];
    const float* q1 = us + 3u * (unsigned)n[1];
    const float* q2 = us + 3u * (unsigned)n[2];
    const float* q3 = us + 3u * (unsigned)n[3];
    float u0x = q0[0], u0y = q0[1], u0z = q0[2];
    float d1x = q1[0] - u0x, d1y = q1[1] - u0y, d1z = q1[2] - u0z;
    float d2x = q2[0] - u0x, d2y = q2[1] - u0y, d2z = q2[2] - u0z;
    float d3x = q3[0] - u0x, d3y = q3[1] - u0y, d3z = q3[2] - u0z;

    // H[i][j] = grad_u = d1[i]*r0[j] + d2[i]*r1[j] + d3[i]*r2[j]
    float H00 = d1x*r0x + d2x*r1x + d3x*r2x;
    float H01 = d1x*r0y + d2x*r1y + d3x*r2y;
    float H02 = d1x*r0z + d2x*r1z + d3x*r2z;
    float H10 = d1y*r0x + d2y*r1x + d3y*r2x;
    float H11 = d1y*r0y + d2y*r1y + d3y*r2y;
    float H12 = d1y*r0z + d2y*r1z + d3y*r2z;
    float H20 = d1z*r0x + d2z*r1x + d3z*r2x;
    float H21 = d1z*r0y + d2z*r1y + d3z*r2y;
    float H22 = d1z*r0z + d2z*r1z + d3z*r2z;

    float tr  = H00 + H11 + H22;
    float e01 = H01 + H10, e02 = H02 + H20, e12 = H12 + H21;
    // sum(eps*eps) = diag^2 + 0.5*(Hij+Hji)^2  (off-diag pairs)
    float epsF2 = H00*H00 + H11*H11 + H22*H22
                + 0.5f * (e01*e01 + e02*e02 + e12*e12);
    // psi = MU*epsF2 + 0.5*LAM*tr^2  with MU=0.5, LAM=1.0
    float psi = 0.5f * epsF2 + 0.5f * tr * tr;
    return det * (1.0f / 6.0f) * psi;       // JxW * psi
}

__global__ void __launch_bounds__(BLOCK)
tet_energy_partials(const float* __restrict__ coords,
                    const float* __restrict__ us,
                    const int*   __restrict__ conns,
                    float* __restrict__ partials, int ne) {
    float acc0 = 0.0f, acc1 = 0.0f;
    int tid    = blockIdx.x * BLOCK + threadIdx.x;
    int stride = gridDim.x * BLOCK;
    for (int e = tid; e < ne; e += 2 * stride) {
        // Prefetch next iteration's index quad (global_prefetch_b8);
        // clamped address keeps EXEC uniform.
        int epf = e + 2 * stride;
        epf = epf < ne ? epf : e;
        __builtin_prefetch(conns + 4u * (unsigned)epf, 0, 0);

        acc0 += elem_energy(e, coords, us, conns);
        int e2 = e + stride;
        if (e2 < ne) acc1 += elem_energy(e2, coords, us, conns);
    }

    float wsum = wave_sum_wmma(acc0, acc1);   // EXEC all-ones here

    __shared__ float lds[BLOCK / WAVE];
    int wave = threadIdx.x / WAVE;
    int lane = threadIdx.x % WAVE;
    if (lane == 0) lds[wave] = wsum;
    __syncthreads();
    if (threadIdx.x == 0) {
        float bsum = 0.0f;
#pragma unroll
        for (int i = 0; i < BLOCK / WAVE; ++i) bsum += lds[i];
        partials[blockIdx.x] = bsum;
    }
}

__global__ void __launch_bounds__(BLOCK)
reduce_partials(const float* __restrict__ partials, int n,
                float* __restrict__ out) {
    float acc0 = 0.0f, acc1 = 0.0f;
    for (int i = threadIdx.x; i < n; i += 2 * BLOCK) {
        acc0 += partials[i];
        int j = i + BLOCK;
        if (j < n) acc1 += partials[j];
    }
    float wsum = wave_sum_wmma(acc0, acc1);

    __shared__ float lds[BLOCK / WAVE];
    int wave = threadIdx.x / WAVE;
    int lane = threadIdx.x % WAVE;
    if (lane == 0) lds[wave] = wsum;
    __syncthreads();
    if (threadIdx.x == 0) {
        float t = 0.0f;
#pragma unroll
        for (int i = 0; i < BLOCK / WAVE; ++i) t += lds[i];
        out[0] = t;
    }
}

// state passthrough: 4 floats/thread, NT both directions (pure stream).
// src is 16B aligned (base of state_old); dst = out+1 is only 4B aligned,
// so store as four b32 NT stores.
__global__ void __launch_bounds__(BLOCK)
copy_state(const float* __restrict__ src, float* __restrict__ dst, int n) {
    int i4 = (blockIdx.x * BLOCK + threadIdx.x) * 4;
    if (i4 + 3 < n) {
        v4f v = __builtin_nontemporal_load((const v4f*)(src + i4));
        __builtin_nontemporal_store(v[0], dst + i4 + 0);
        __builtin_nontemporal_store(v[1], dst + i4 + 1);
        __builtin_nontemporal_store(v[2], dst + i4 + 2);
        __builtin_nontemporal_store(v[3], dst + i4 + 3);
    } else {
        for (int i = i4; i < n; ++i) dst[i] = src[i];
    }
}

// ---------------------------------------------------------------------------
// Inputs: 0=coords(f32), 1=us(f32), 2=conns(i32), 3=t(f32, unused),
//         4=state_old(f32), 5=dt(f32, unused)
// Output: [pi (1 float), state_new (N_ELS floats)]
// ---------------------------------------------------------------------------
extern "C" void kernel_launch(void* const* d_in, const int* in_sizes, int n_in,
                              void* d_out, int out_size, void* d_ws, size_t ws_size,
                              hipStream_t stream) {
    const float* coords    = (const float*)d_in[0];
    const float* us        = (const float*)d_in[1];
    const int*   conns     = (const int*)d_in[2];
    const float* state_old = (const float*)d_in[4];
    float*       out       = (float*)d_out;

    int ne     = in_sizes[2] / 4;     // elements
    int nstate = in_sizes[4];         // state passthrough count
    float* partials = (float*)d_ws;   // NBLOCKS floats of scratch

    tet_energy_partials<<<NBLOCKS, BLOCK, 0, stream>>>(coords, us, conns,
                                                       partials, ne);
    reduce_partials<<<1, BLOCK, 0, stream>>>(partials, NBLOCKS, out);

    int nq = (nstate + 3) / 4;
    copy_state<<<(nq + BLOCK - 1) / BLOCK, BLOCK, 0, stream>>>(
        state_old, out + 1, nstate);
}